// PrototypeDynamicCalibrator_45947560132957
// MI455X (gfx1250) — compile-verified
//
#include <hip/hip_runtime.h>
#include <math.h>

typedef float v2f __attribute__((ext_vector_type(2)));
typedef float v8f __attribute__((ext_vector_type(8)));

#define Bn 1024
#define Tn 64
#define Dn 256
#define Pn 2048
#define Hn 512

// ---------------------------------------------------------------------------
// Single-wave WMMA f32 GEMM, 16(M) x 32(N) C-tile per wave (two accumulators
// sharing one A fragment). TRANSB/MODE are compile-time so the K-loop is just
// load-b64 / load-b64 / load-b64 / wmma / wmma with pointer bumps.
// MODE: 0 = +bias, 1 = +bias,ReLU, 2 = drift epilogue 1 - acc/(fn[m]*pn[n])
// TRANSB: 0 -> B[K,N] row-major, 1 -> B[N,K] row-major (NT gemm)
// ---------------------------------------------------------------------------
template<int TRANSB, int MODE>
__global__ __launch_bounds__(32)
void wmma_gemm_f32(const float* __restrict__ A, int lda,
                   const float* __restrict__ Bm, int ldb,
                   const float* __restrict__ bias,
                   float* __restrict__ C, int ldc,
                   int K,
                   const float* __restrict__ fn, const float* __restrict__ pn)
{
    const int lane = threadIdx.x;        // EXEC all ones (WMMA requirement)
    const int n0   = blockIdx.x * 32;
    const int m0   = blockIdx.y * 16;
    const int half = lane >> 4;          // ISA f32 frag layout: lanes 16-31 hold K+2,K+3
    const int l    = lane & 15;

    const float* aptr = A + (unsigned)((m0 + l) * lda + half * 2);
    const float* bptr0;
    const float* bptr1;
    int bstep;
    if (TRANSB) {
        bptr0 = Bm + (unsigned)((n0 + l) * ldb + half * 2);
        bptr1 = bptr0 + 16 * ldb;
        bstep = 4;
    } else {
        bptr0 = Bm + (unsigned)(half * 2 * ldb + n0 + l);
        bptr1 = bptr0 + 16;
        bstep = 4 * ldb;
    }

    v8f acc0 = {}, acc1 = {};
    #pragma unroll 4
    for (int k = 0; k < K; k += 4) {
        v2f a = *(const v2f*)aptr;                 // global_load_b64
        v2f b0, b1;
        if (TRANSB) {
            b0 = *(const v2f*)bptr0;               // global_load_b64
            b1 = *(const v2f*)bptr1;
        } else {
            b0.x = bptr0[0];  b0.y = bptr0[ldb];
            b1.x = bptr1[0];  b1.y = bptr1[ldb];
        }
        acc0 = __builtin_amdgcn_wmma_f32_16x16x4_f32(
            false, a, false, b0, (short)0, acc0, false, false);
        acc1 = __builtin_amdgcn_wmma_f32_16x16x4_f32(
            false, a, false, b1, (short)0, acc1, false, false);
        aptr  += 4;
        bptr0 += bstep;
        bptr1 += bstep;
    }

    #pragma unroll
    for (int t = 0; t < 2; ++t) {
        const v8f acc = t ? acc1 : acc0;
        const int col = n0 + t * 16 + l;
        float bv = 0.0f, pv = 0.0f;
        if (MODE == 2) pv = pn[col]; else bv = bias[col];
        #pragma unroll
        for (int r = 0; r < 8; ++r) {
            const int row = m0 + r + half * 8;     // ISA C/D: VGPR r -> M=r (+8 hi half)
            float v = acc[r];
            if (MODE == 2) {
                v = 1.0f - v / (fn[row] * pv);
            } else {
                v += bv;
                if (MODE == 1) v = fmaxf(v, 0.0f);
            }
            C[(size_t)row * ldc + col] = v;
        }
    }
}

// mean over T axis: X[R,T,D] -> Y[R,D]
__global__ void mean_over_T_kernel(const float* __restrict__ X, float* __restrict__ Y, int R)
{
    int idx = blockIdx.x * blockDim.x + threadIdx.x;   // r*D + d
    if (idx >= R * Dn) return;
    int r = idx >> 8, d = idx & (Dn - 1);
    const float* p = X + (size_t)r * Tn * Dn + d;
    float s = 0.0f;
    #pragma unroll 4
    for (int t = 0; t < Tn; ++t) s += p[(size_t)t * Dn];
    Y[idx] = s * (1.0f / Tn);
}

// per-row L2 norm, clamped at EPS_COS. One 256-thread block per row (D=256).
__global__ void row_norm_kernel(const float* __restrict__ X, float* __restrict__ nrm)
{
    __shared__ float red[256];
    const int row = blockIdx.x;
    float v = X[(size_t)row * Dn + threadIdx.x];
    red[threadIdx.x] = v * v;
    __syncthreads();
    for (int off = 128; off > 0; off >>= 1) {
        if (threadIdx.x < off) red[threadIdx.x] += red[threadIdx.x + off];
        __syncthreads();
    }
    if (threadIdx.x == 0) nrm[row] = fmaxf(sqrtf(red[0]), 1e-8f);
}

// column mean over rows: X[R,D] -> out[D]
__global__ void col_mean_kernel(const float* __restrict__ X, float* __restrict__ out, int R)
{
    int d = blockIdx.x * blockDim.x + threadIdx.x;
    if (d >= Dn) return;
    float s = 0.0f;
    for (int r = 0; r < R; ++r) s += X[(size_t)r * Dn + d];
    out[d] = s / (float)R;
}

// concat[p, 0:256] = pc[p], concat[p, 256:512] = gmean
__global__ void build_concat_kernel(const float* __restrict__ pc,
                                    const float* __restrict__ gmean,
                                    float* __restrict__ cc)
{
    int idx = blockIdx.x * blockDim.x + threadIdx.x;
    if (idx >= Pn * 2 * Dn) return;
    int p = idx >> 9, j = idx & 511;
    cc[idx] = (j < Dn) ? pc[(size_t)p * Dn + j] : gmean[j - Dn];
}

// drift_mean over B, threshold -> int flag + float flag output
__global__ void drift_reduce_kernel(const float* __restrict__ ds,
                                    int* __restrict__ drifted,
                                    float* __restrict__ flag_out)
{
    int p = blockIdx.x * blockDim.x + threadIdx.x;
    if (p >= Pn) return;
    float s = 0.0f;
    for (int b = 0; b < Bn; ++b) s += ds[(size_t)b * Pn + p];
    int dr = (s / (float)Bn) > 0.3f;
    drifted[p]  = dr;
    flag_out[p] = dr ? 1.0f : 0.0f;
}

// in-place LayerNorm (+ g,b affine) + ReLU over last dim Hc. Block per row.
__global__ void ln_relu_kernel(float* __restrict__ X, int Hc,
                               const float* __restrict__ g, const float* __restrict__ be)
{
    __shared__ float red[256];
    float* x = X + (size_t)blockIdx.x * Hc;

    float s = 0.0f;
    for (int i = threadIdx.x; i < Hc; i += 256) s += x[i];
    red[threadIdx.x] = s;
    __syncthreads();
    for (int off = 128; off > 0; off >>= 1) {
        if (threadIdx.x < off) red[threadIdx.x] += red[threadIdx.x + off];
        __syncthreads();
    }
    const float mu = red[0] / (float)Hc;
    __syncthreads();

    float v = 0.0f;
    for (int i = threadIdx.x; i < Hc; i += 256) { float d = x[i] - mu; v += d * d; }
    red[threadIdx.x] = v;
    __syncthreads();
    for (int off = 128; off > 0; off >>= 1) {
        if (threadIdx.x < off) red[threadIdx.x] += red[threadIdx.x + off];
        __syncthreads();
    }
    const float rstd = rsqrtf(red[0] / (float)Hc + 1e-5f);
    __syncthreads();

    for (int i = threadIdx.x; i < Hc; i += 256) {
        float y = (x[i] - mu) * rstd * g[i] + be[i];
        x[i] = fmaxf(y, 0.0f);
    }
}

// conf = sigmoid(h2[p,:] . w3 + b3); strength = clip(0.1*conf, 0, 0.5)
__global__ void conf_strength_kernel(const float* __restrict__ h2,
                                     const float* __restrict__ w3,
                                     const float* __restrict__ b3,
                                     float* __restrict__ strength)
{
    int p = blockIdx.x * blockDim.x + threadIdx.x;
    if (p >= Pn) return;
    const float* r = h2 + (size_t)p * (Hn / 2);
    float s = b3[0];
    for (int i = 0; i < Hn / 2; ++i) s += r[i] * w3[i];
    float conf = 1.0f / (1.0f + expf(-s));
    strength[p] = fminf(fmaxf(0.1f * conf, 0.0f), 0.5f);
}

// calibrated[p,t,d] = drifted[p] ? pc[p,d] + strength[p]*delta[p,d] : prototype[p,t,d]
__global__ void calibrate_kernel(const float* __restrict__ prototype,
                                 const float* __restrict__ pc,
                                 const float* __restrict__ delta,
                                 const float* __restrict__ strength,
                                 const int* __restrict__ drifted,
                                 float* __restrict__ out)
{
    size_t idx = (size_t)blockIdx.x * blockDim.x + threadIdx.x;
    if (idx >= (size_t)Pn * Tn * Dn) return;
    int p = (int)(idx / (Tn * Dn));
    int d = (int)(idx & (Dn - 1));
    if (drifted[p])
        out[idx] = pc[(size_t)p * Dn + d] + strength[p] * delta[(size_t)p * Dn + d];
    else
        out[idx] = prototype[idx];
}

extern "C" void kernel_launch(void* const* d_in, const int* in_sizes, int n_in,
                              void* d_out, int out_size, void* d_ws, size_t ws_size,
                              hipStream_t stream) {
    (void)in_sizes; (void)n_in; (void)out_size; (void)ws_size;

    const float* current_feat = (const float*)d_in[0];
    const float* prototype    = (const float*)d_in[1];
    const float* dd_w1  = (const float*)d_in[2];
    const float* dd_b1  = (const float*)d_in[3];
    const float* dd_g1  = (const float*)d_in[4];
    const float* dd_be1 = (const float*)d_in[5];
    const float* dd_w2  = (const float*)d_in[6];
    const float* dd_b2  = (const float*)d_in[7];
    const float* dd_w3  = (const float*)d_in[8];
    const float* dd_b3  = (const float*)d_in[9];
    const float* cal_w1  = (const float*)d_in[10];
    const float* cal_b1  = (const float*)d_in[11];
    const float* cal_g1  = (const float*)d_in[12];
    const float* cal_be1 = (const float*)d_in[13];
    const float* cal_w2  = (const float*)d_in[14];
    const float* cal_b2  = (const float*)d_in[15];
    const float* cal_w3  = (const float*)d_in[16];
    const float* cal_b3  = (const float*)d_in[17];

    // output layout: calibrated [P,T,D] | drift_score [B,P] | is_drifted [P]
    float* out_calib = (float*)d_out;
    float* out_drift = out_calib + (size_t)Pn * Tn * Dn;
    float* out_flag  = out_drift + (size_t)Bn * Pn;

    // workspace layout (floats)
    float* ws = (float*)d_ws;
    float* feat_flat = ws;                                 // [B,D]
    float* pc        = feat_flat + (size_t)Bn * Dn;        // [P,D]
    float* fn        = pc + (size_t)Pn * Dn;               // [B]
    float* pn        = fn + Bn;                            // [P]
    float* gmean     = pn + Pn;                            // [D]
    float* bufA      = gmean + Dn;                         // [P,H]
    float* bufB      = bufA + (size_t)Pn * Hn;             // [P,2D] / [P,H] (concat alias)
    float* delta     = bufB + (size_t)Pn * Hn;             // [P,D]
    float* strength  = delta + (size_t)Pn * Dn;            // [P]
    int*   drifted   = (int*)(strength + Pn);              // [P]

    // 1) T-mean reductions
    mean_over_T_kernel<<<(Bn * Dn) / 256, 256, 0, stream>>>(current_feat, feat_flat, Bn);
    mean_over_T_kernel<<<(Pn * Dn) / 256, 256, 0, stream>>>(prototype, pc, Pn);

    // 2) norms + global mean
    row_norm_kernel<<<Bn, 256, 0, stream>>>(feat_flat, fn);
    row_norm_kernel<<<Pn, 256, 0, stream>>>(pc, pn);
    col_mean_kernel<<<1, 256, 0, stream>>>(feat_flat, gmean, Bn);

    // 3) drift_score = 1 - cos-sim  (WMMA NT gemm, drift epilogue)
    wmma_gemm_f32<1, 2><<<dim3(Pn / 32, Bn / 16), 32, 0, stream>>>(
        feat_flat, Dn, pc, Dn, nullptr, out_drift, Pn, Dn, fn, pn);

    // 4) drifted flags
    drift_reduce_kernel<<<Pn / 256, 256, 0, stream>>>(out_drift, drifted, out_flag);

    // 5) drift-detector MLP
    build_concat_kernel<<<(Pn * 2 * Dn) / 256, 256, 0, stream>>>(pc, gmean, bufB);
    wmma_gemm_f32<0, 0><<<dim3(Hn / 32, Pn / 16), 32, 0, stream>>>(     // concat @ dd_w1 + b1
        bufB, 2 * Dn, dd_w1, Hn, dd_b1, bufA, Hn, 2 * Dn, nullptr, nullptr);
    ln_relu_kernel<<<Pn, 256, 0, stream>>>(bufA, Hn, dd_g1, dd_be1);
    wmma_gemm_f32<0, 1><<<dim3((Hn / 2) / 32, Pn / 16), 32, 0, stream>>>( // h1 @ dd_w2, relu
        bufA, Hn, dd_w2, Hn / 2, dd_b2, bufB, Hn / 2, Hn, nullptr, nullptr);
    conf_strength_kernel<<<Pn / 256, 256, 0, stream>>>(bufB, dd_w3, dd_b3, strength);

    // 6) calibration MLP
    wmma_gemm_f32<0, 0><<<dim3(Hn / 32, Pn / 16), 32, 0, stream>>>(     // pc @ cal_w1 + b1
        pc, Dn, cal_w1, Hn, cal_b1, bufA, Hn, Dn, nullptr, nullptr);
    ln_relu_kernel<<<Pn, 256, 0, stream>>>(bufA, Hn, cal_g1, cal_be1);
    wmma_gemm_f32<0, 1><<<dim3(Hn / 32, Pn / 16), 32, 0, stream>>>(     // c1 @ cal_w2, relu
        bufA, Hn, cal_w2, Hn, cal_b2, bufB, Hn, Hn, nullptr, nullptr);
    wmma_gemm_f32<0, 0><<<dim3(Dn / 32, Pn / 16), 32, 0, stream>>>(     // c2 @ cal_w3 + b3
        bufB, Hn, cal_w3, Dn, cal_b3, delta, Dn, Hn, nullptr, nullptr);

    // 7) final masked scatter into calibrated [P,T,D]
    calibrate_kernel<<<(int)(((size_t)Pn * Tn * Dn) / 256), 256, 0, stream>>>(
        prototype, pc, delta, strength, drifted, out_calib);
}